// Dice_824633721227
// MI455X (gfx1250) — compile-verified
//
#include <hip/hip_runtime.h>
#include <stdint.h>

// ---------------- problem constants (from reference) ----------------
#define BATCH   4
#define NCLASS  5
#define NPOS    (128*128*128)        // 2,097,152 voxels per batch
#define CHUNK   1024                 // voxels per TDM tile (divides NPOS)
#define NCHUNK  (NPOS / CHUNK)       // 2048 tiles per batch
#define BLOCKS_X 512
#define THREADS 256                  // 8 wave32 per block
#define TILE_FLOATS (NCLASS * CHUNK) // 5120 floats = 20 KB
#define TILE_BYTES  (TILE_FLOATS * 4)

typedef unsigned int u32x4 __attribute__((ext_vector_type(4)));
typedef int          i32x8 __attribute__((ext_vector_type(8)));
typedef int          i32x4 __attribute__((ext_vector_type(4)));
typedef int          vi4   __attribute__((ext_vector_type(4)));
typedef float        vf4   __attribute__((ext_vector_type(4)));

// Issue one TDM DMA: a 2D tile (tile_dim0=CHUNK floats, tile_dim1=NCLASS rows,
// row stride = NPOS floats) from global memory into LDS at lds_byte_off.
// D# layout per CDNA5 ISA.
__device__ __forceinline__ void tdm_issue_tile(const float* gsrc, uint32_t lds_byte_off) {
    uint64_t ga = (uint64_t)(uintptr_t)gsrc;
    u32x4 g0;
    g0[0] = 1u;                                            // count=1, user mode
    g0[1] = lds_byte_off;                                  // lds_addr
    g0[2] = (uint32_t)(ga & 0xFFFFFFFFu);                  // global_addr[31:0]
    g0[3] = (uint32_t)((ga >> 32) & 0x01FFFFFFu)           // global_addr[56:32]
          | (2u << 30);                                    // type=2 ("image")
    i32x8 g1;
    g1[0] = 0x00020000;                                    // data_size=2 (4B); no multicast/pad
    g1[1] = (int)((CHUNK & 0xFFFF) << 16);                 // tensor_dim0[15:0] @bits63:48
    g1[2] = (int)(((CHUNK >> 16) & 0xFFFF)                 // tensor_dim0[31:16]
          | ((NCLASS & 0xFFFF) << 16));                    // tensor_dim1[15:0]
    g1[3] = (int)((CHUNK & 0xFFFF) << 16);                 // tensor_dim1[31:16]=0 | tile_dim0
    g1[4] = NCLASS;                                        // tile_dim1 | tile_dim2(0)<<16
    g1[5] = NPOS;                                          // tensor_dim0_stride[31:0] (elements)
    g1[6] = 0;                                             // stride0[47:32] | stride1[15:0]
    g1[7] = 0;                                             // stride1[47:16]
    i32x4 gz = {0, 0, 0, 0};
#if __clang_major__ >= 23
    i32x8 gz8 = {0, 0, 0, 0, 0, 0, 0, 0};
    __builtin_amdgcn_tensor_load_to_lds(g0, g1, gz, gz, gz8, 0);
#else
    __builtin_amdgcn_tensor_load_to_lds(g0, g1, gz, gz, 0);
#endif
}

__global__ void __launch_bounds__(THREADS)
dice_count_kernel(const float* __restrict__ inp,
                  const long long* __restrict__ tgt,
                  int* __restrict__ cnt /* [BATCH][8]: union c1..c4, inter c1..c4 */) {
    __shared__ __align__(16) float tile[2][TILE_FLOATS];   // double-buffered TDM dest, LDS off 0
    __shared__ int s_cnt[8];

    const int b   = blockIdx.y;
    const int tid = threadIdx.x;
    if (tid < 8) s_cnt[tid] = 0;

    const float*     base  = inp + (size_t)b * NCLASS * NPOS;
    const long long* tbase = tgt + (size_t)b * NPOS;

    int u1 = 0, u2 = 0, u3 = 0, u4 = 0;   // pred_c + tgt_c  (classes 1..4)
    int n1 = 0, n2 = 0, n3 = 0, n4 = 0;   // intersection    (classes 1..4)

    // Wave-uniform scalar guard: only wave 0 drives the TDM (TDM ignores EXEC,
    // so a scalar branch is required to keep other waves from issuing it).
    const bool wave0 = (__builtin_amdgcn_readfirstlane(threadIdx.x) >> 5) == 0;

    int cid = blockIdx.x;
    if (wave0) tdm_issue_tile(base + (size_t)cid * CHUNK, 0);   // prologue: tile 0 -> buf 0
    __syncthreads();                                            // covers s_cnt init too

    int buf = 0;
    for (; cid < NCHUNK; cid += gridDim.x) {
        const int nxt = cid + gridDim.x;
        if (wave0) {
            if (nxt < NCHUNK) {
                tdm_issue_tile(base + (size_t)nxt * CHUNK, (uint32_t)((buf ^ 1) * TILE_BYTES));
                __builtin_amdgcn_s_wait_tensorcnt(1);   // oldest (current tile) complete
            } else {
                __builtin_amdgcn_s_wait_tensorcnt(0);
            }
        }
        __syncthreads();        // current tile visible to all waves

        // ---- compute on tile[buf]: 4 voxels per thread (256*4 == CHUNK) ----
        const int p = tid * 4;
        vf4 v0 = *reinterpret_cast<const vf4*>(&tile[buf][0 * CHUNK + p]);
        vf4 v1 = *reinterpret_cast<const vf4*>(&tile[buf][1 * CHUNK + p]);
        vf4 v2 = *reinterpret_cast<const vf4*>(&tile[buf][2 * CHUNK + p]);
        vf4 v3 = *reinterpret_cast<const vf4*>(&tile[buf][3 * CHUNK + p]);
        vf4 v4 = *reinterpret_cast<const vf4*>(&tile[buf][4 * CHUNK + p]);

        // target: 4 x int64, stream with non-temporal 128-bit loads
        const vi4* t4 = reinterpret_cast<const vi4*>(tbase + (size_t)cid * CHUNK + p);
        vi4 ta = __builtin_nontemporal_load(t4);
        vi4 tb = __builtin_nontemporal_load(t4 + 1);
        int tv[4] = { ta[0], ta[2], tb[0], tb[2] };   // low dwords (labels 0..4)

        #pragma unroll
        for (int k = 0; k < 4; ++k) {
            float best = v0[k];
            int pred = 0;
            if (v1[k] > best) { best = v1[k]; pred = 1; }   // strict > : first-max index
            if (v2[k] > best) { best = v2[k]; pred = 2; }
            if (v3[k] > best) { best = v3[k]; pred = 3; }
            if (v4[k] > best) { best = v4[k]; pred = 4; }
            const int t = tv[k];
            u1 += (pred == 1) + (t == 1);  n1 += (pred == 1) & (t == 1);
            u2 += (pred == 2) + (t == 2);  n2 += (pred == 2) & (t == 2);
            u3 += (pred == 3) + (t == 3);  n3 += (pred == 3) & (t == 3);
            u4 += (pred == 4) + (t == 4);  n4 += (pred == 4) & (t == 4);
        }
        __syncthreads();        // tile[buf] free before it is DMA-overwritten
        buf ^= 1;
    }

    // ---- block reduction ----
    atomicAdd(&s_cnt[0], u1); atomicAdd(&s_cnt[1], u2);
    atomicAdd(&s_cnt[2], u3); atomicAdd(&s_cnt[3], u4);
    atomicAdd(&s_cnt[4], n1); atomicAdd(&s_cnt[5], n2);
    atomicAdd(&s_cnt[6], n3); atomicAdd(&s_cnt[7], n4);
    __syncthreads();
    if (tid < 8) atomicAdd(&cnt[b * 8 + tid], s_cnt[tid]);
}

__global__ void dice_final_kernel(const int* __restrict__ cnt, float* __restrict__ out) {
    if (threadIdx.x == 0 && blockIdx.x == 0) {
        float acc = 0.0f;
        #pragma unroll
        for (int b = 0; b < BATCH; ++b) {
            #pragma unroll
            for (int c = 0; c < 4; ++c) {
                const float uni   = (float)cnt[b * 8 + c];
                const float inter = (float)cnt[b * 8 + 4 + c];
                acc += (2.0f * inter + 1e-5f) / (uni + 1e-5f);
            }
        }
        out[0] = acc / 16.0f;   // mean over batch then classes == mean over all 16
    }
}

extern "C" void kernel_launch(void* const* d_in, const int* in_sizes, int n_in,
                              void* d_out, int out_size, void* d_ws, size_t ws_size,
                              hipStream_t stream) {
    const float*     inp = (const float*)d_in[0];
    const long long* tgt = (const long long*)d_in[1];
    float*           out = (float*)d_out;
    int*             cnt = (int*)d_ws;

    hipMemsetAsync(cnt, 0, BATCH * 8 * sizeof(int), stream);   // graph-capturable
    dim3 grid(BLOCKS_X, BATCH);
    dice_count_kernel<<<grid, THREADS, 0, stream>>>(inp, tgt, cnt);
    dice_final_kernel<<<1, 1, 0, stream>>>(cnt, out);
    (void)in_sizes; (void)n_in; (void)out_size; (void)ws_size;
}